// denoise_Pool_v2_52278341927163
// MI455X (gfx1250) — compile-verified
//
#include <hip/hip_runtime.h>
#include <hip/hip_bf16.h>

// ---------------------------------------------------------------------------
// CDNA5 (gfx1250) implementation of the graph-denoise pipeline.
//   N = 4096 rows, D = 256 features (D hardcoded; reference fixes D=256).
//   GEMM h = feat @ W^T + b uses V_WMMA_F32_16X16X4_F32 (exact fp32 path).
// ---------------------------------------------------------------------------

typedef __attribute__((ext_vector_type(2))) float v2f;
typedef __attribute__((ext_vector_type(8))) float v8f;

#define FEAT_D 256
#define NBINS 4096
#define LDS_STRIDE 260   // 256 + 4 pad to dodge bank conflicts

// --- Kernel 1: fused GEMM (WMMA f32 16x16x4) + bias + LeakyReLU + row L2 norm
__global__ __launch_bounds__(256) void gemm_norm_kernel(
    const float* __restrict__ feat, const float* __restrict__ W,
    const float* __restrict__ bias, const float* __restrict__ alpha,
    float* __restrict__ feat_norm)
{
    __shared__ float h_s[16 * LDS_STRIDE];
    __shared__ float sumsq[16];

    const int t    = threadIdx.x;
    const int wave = t >> 5;
    const int lane = t & 31;
    const int hlf  = lane >> 4;   // which half of the wave (K pair select)
    const int lr   = lane & 15;
    const int rowBase = blockIdx.x * 16;

    if (t < 16) sumsq[t] = 0.0f;

    // Each wave owns two 16-wide column tiles: [n0, n0+16) and [n0+16, n0+32)
    const int n0 = wave * 32;
    v8f acc0 = {};
    v8f acc1 = {};

    // A (16x4, M x K): lanes 0-15 hold M=lane, K={k,k+1}; lanes 16-31 M=lane-16, K={k+2,k+3}
    // B (4x16, K x N): B[k][n] = W[n][k]; mirrored layout -> lane loads W[n0+lr][k + hlf*2 .. +1]
    const float* arow  = feat + (size_t)(rowBase + lr) * FEAT_D + hlf * 2;
    const float* b0row = W    + (size_t)(n0 + lr)      * FEAT_D + hlf * 2;
    const float* b1row = W    + (size_t)(n0 + 16 + lr) * FEAT_D + hlf * 2;

    for (int k = 0; k < FEAT_D; k += 4) {
        v2f a  = *(const v2f*)(arow  + k);
        v2f b0 = *(const v2f*)(b0row + k);
        v2f b1 = *(const v2f*)(b1row + k);
        acc0 = __builtin_amdgcn_wmma_f32_16x16x4_f32(false, a, false, b0,
                                                     (short)0, acc0, false, false);
        acc1 = __builtin_amdgcn_wmma_f32_16x16x4_f32(false, a, false, b1,
                                                     (short)0, acc1, false, false);
    }

    // C/D layout: VGPR v -> row v (lanes 0-15) / row v+8 (lanes 16-31); col = tile_base + (lane&15)
    const int c0 = n0 + lr;
    const int c1 = n0 + 16 + lr;
    const float bias0 = bias[c0],  bias1 = bias[c1];
    const float alp0  = alpha[c0], alp1  = alpha[c1];
#pragma unroll
    for (int v = 0; v < 8; ++v) {
        const int r = v + hlf * 8;
        float x0 = acc0[v] + bias0;
        x0 = (x0 >= 0.0f) ? x0 : alp0 * x0;
        h_s[r * LDS_STRIDE + c0] = x0;
        float x1 = acc1[v] + bias1;
        x1 = (x1 >= 0.0f) ? x1 : alp1 * x1;
        h_s[r * LDS_STRIDE + c1] = x1;
    }
    __syncthreads();

    // Row sum-of-squares: 16 threads per row, 16-col chunks each
    {
        const int r  = t & 15;
        const int cc = (t >> 4) * 16;
        float s = 0.0f;
#pragma unroll
        for (int j = 0; j < 16; ++j) {
            const float x = h_s[r * LDS_STRIDE + cc + j];
            s += x * x;
        }
        atomicAdd(&sumsq[r], s);
    }
    __syncthreads();

    // Scale and emit feat_norm
    {
        const int r  = t & 15;
        const int cc = (t >> 4) * 16;
        const float inv = 1.0f / fmaxf(sqrtf(sumsq[r]), 1e-12f);
        float* outp = feat_norm + (size_t)(rowBase + r) * FEAT_D + cc;
#pragma unroll
        for (int j = 0; j < 16; ++j)
            outp[j] = h_s[r * LDS_STRIDE + cc + j] * inv;
    }
}

__device__ __forceinline__ int sim_bin(float s) {
    int b = (int)(s * (float)NBINS);
    return b < 0 ? 0 : (b > NBINS - 1 ? NBINS - 1 : b);
}

// --- Kernel 2: one wave32 per edge: cosine sim + fused histogram
__global__ __launch_bounds__(256) void edge_sim_kernel(
    const int* __restrict__ pair, const float* __restrict__ fn,
    float* __restrict__ sim_out, unsigned* __restrict__ bins, int E)
{
    const int gwave = (int)((blockIdx.x * blockDim.x + threadIdx.x) >> 5);
    const int lane  = threadIdx.x & 31;
    if (gwave >= E) return;
    const int src = pair[2 * gwave];
    const int dst = pair[2 * gwave + 1];
    const float4* ps = (const float4*)(fn + (size_t)src * FEAT_D) + lane * 2;
    const float4* pd = (const float4*)(fn + (size_t)dst * FEAT_D) + lane * 2;
    const float4 s0 = ps[0], s1 = ps[1];
    const float4 d0 = pd[0], d1 = pd[1];
    float dot = s0.x * d0.x + s0.y * d0.y + s0.z * d0.z + s0.w * d0.w
              + s1.x * d1.x + s1.y * d1.y + s1.z * d1.z + s1.w * d1.w;
    for (int off = 16; off > 0; off >>= 1)
        dot += __shfl_down(dot, off, 32);
    if (lane == 0) {
        const float sim = (dot + 1.0f) * 0.5f;
        sim_out[gwave] = sim;
        atomicAdd(&bins[sim_bin(sim)], 1u);
    }
}

// --- Kernel 3: zero scratch
__global__ void zero_ws_kernel(unsigned* __restrict__ bins, unsigned* __restrict__ offs,
                               float* __restrict__ removedsum, unsigned* __restrict__ nnz,
                               int N)
{
    const int i = blockIdx.x * blockDim.x + threadIdx.x;
    if (i < NBINS) { bins[i] = 0u; offs[i] = 0u; }
    if (i < N) removedsum[i] = 0.0f;
    if (i == 0) *nnz = 0u;
}

// --- Kernel 4: single-block exclusive scan of NBINS histogram (in place)
__global__ __launch_bounds__(256) void scan_kernel(unsigned* __restrict__ bins)
{
    __shared__ unsigned partial[256];
    const int t = threadIdx.x;
    unsigned v[NBINS / 256];
    unsigned s = 0;
#pragma unroll
    for (int j = 0; j < NBINS / 256; ++j) { v[j] = bins[t * (NBINS / 256) + j]; s += v[j]; }
    partial[t] = s;
    __syncthreads();
    if (t == 0) {
        unsigned run = 0;
        for (int i = 0; i < 256; ++i) { const unsigned c = partial[i]; partial[i] = run; run += c; }
    }
    __syncthreads();
    unsigned run = partial[t];
#pragma unroll
    for (int j = 0; j < NBINS / 256; ++j) {
        const unsigned c = v[j];
        bins[t * (NBINS / 256) + j] = run;
        run += c;
    }
}

// --- Kernel 5: counting-sort scatter -> order[]
__global__ void scatter_kernel(const float* __restrict__ sim, const unsigned* __restrict__ prefix,
                               unsigned* __restrict__ offs, unsigned* __restrict__ order, int E)
{
    const int e = blockIdx.x * blockDim.x + threadIdx.x;
    if (e >= E) return;
    const int b = sim_bin(sim[e]);
    const unsigned pos = prefix[b] + atomicAdd(&offs[b], 1u);
    order[pos] = (unsigned)e;
}

// --- Kernel 6: per-row sum + global nonzero count (one block per row)
__global__ __launch_bounds__(256) void rowsum_kernel(const float* __restrict__ adj,
                                                     float* __restrict__ rowsum,
                                                     unsigned* __restrict__ nnz, int N)
{
    __shared__ float ssum[256];
    __shared__ unsigned scnt[256];
    const int row = blockIdx.x;
    const float4* a4 = (const float4*)(adj + (size_t)row * N);
    float s = 0.0f;
    unsigned c = 0;
    for (int j = threadIdx.x; j < (N >> 2); j += 256) {
        const float4 x = a4[j];
        s += x.x + x.y + x.z + x.w;
        c += (x.x != 0.0f) + (x.y != 0.0f) + (x.z != 0.0f) + (x.w != 0.0f);
    }
    ssum[threadIdx.x] = s;
    scnt[threadIdx.x] = c;
    __syncthreads();
    for (int off = 128; off > 0; off >>= 1) {
        if (threadIdx.x < off) {
            ssum[threadIdx.x] += ssum[threadIdx.x + off];
            scnt[threadIdx.x] += scnt[threadIdx.x + off];
        }
        __syncthreads();
    }
    if (threadIdx.x == 0) {
        rowsum[row] = ssum[0];
        atomicAdd(nnz, scnt[0]);
    }
}

// --- Kernel 7: accumulate removed weight per source row
__global__ void removed_kernel(const unsigned* __restrict__ order, const int* __restrict__ pair,
                               const float* __restrict__ adj, float* __restrict__ removedsum,
                               int k0, int N)
{
    const int p = blockIdx.x * blockDim.x + threadIdx.x;
    if (p >= k0) return;
    const unsigned e = order[p];
    const int src = pair[2 * e];
    const int dst = pair[2 * e + 1];
    atomicAdd(&removedsum[src], adj[(size_t)src * N + dst]);
}

// --- Kernel 8: streaming normalize new_adj = adj / (deg + 1e-6)
__global__ __launch_bounds__(256) void normalize_kernel(
    const float* __restrict__ adj, const float* __restrict__ rowsum,
    const float* __restrict__ removedsum, float* __restrict__ out, int N)
{
    const size_t total4 = ((size_t)N * (size_t)N) >> 2;
    const size_t stride = (size_t)gridDim.x * blockDim.x;
    const size_t row4   = (size_t)(N >> 2);
    const float4* a4 = (const float4*)adj;
    float4* o4 = (float4*)out;
    for (size_t i = (size_t)blockIdx.x * blockDim.x + threadIdx.x; i < total4; i += stride) {
        const int row = (int)(i / row4);
        const float inv = 1.0f / (rowsum[row] - removedsum[row] + 1e-6f);
        float4 v = a4[i];
        v.x *= inv; v.y *= inv; v.z *= inv; v.w *= inv;
        o4[i] = v;
    }
}

// --- Kernel 9: zero the dropped edges in new_adj
__global__ void zero_removed_kernel(const unsigned* __restrict__ order,
                                    const int* __restrict__ pair,
                                    float* __restrict__ out, int k0, int N)
{
    const int p = blockIdx.x * blockDim.x + threadIdx.x;
    if (p >= k0) return;
    const unsigned e = order[p];
    out[(size_t)pair[2 * e] * N + pair[2 * e + 1]] = 0.0f;
}

// --- Kernel 10: remaining outputs (pair copy, left_row, scalars)
__global__ void finalize_kernel(const int* __restrict__ pair, const unsigned* __restrict__ order,
                                const unsigned* __restrict__ nnz,
                                float* __restrict__ out_pair, float* __restrict__ out_before,
                                float* __restrict__ out_after, float* __restrict__ out_left,
                                int E, int k0, int N)
{
    const int i = blockIdx.x * blockDim.x + threadIdx.x;
    const int npair = 2 * E;
    const int nleft = E - k0;
    if (i < npair) {
        out_pair[i] = (float)pair[i];
    } else if (i < npair + nleft) {
        out_left[i - npair] = (float)order[k0 + (i - npair)];
    } else if (i == npair + nleft) {
        out_before[0] = (float)E / (float)N;
        out_after[0]  = ((float)(*nnz) - (float)k0) / (float)N;
    }
}

extern "C" void kernel_launch(void* const* d_in, const int* in_sizes, int n_in,
                              void* d_out, int out_size, void* d_ws, size_t ws_size,
                              hipStream_t stream)
{
    const float* adj   = (const float*)d_in[0];
    const float* feat  = (const float*)d_in[1];
    const float* W     = (const float*)d_in[2];
    const float* b     = (const float*)d_in[3];
    const float* alpha = (const float*)d_in[4];
    const int*   pair  = (const int*)d_in[5];
    (void)n_in; (void)out_size; (void)ws_size;

    const int Dd = in_sizes[3];              // 256
    const int N  = in_sizes[1] / Dd;         // 4096
    const int E  = in_sizes[5] / 2;
    const int k0 = (int)((double)E * (1.0 - 0.5));

    // Output layout (flat float32, reference tuple order)
    float* out = (float*)d_out;
    size_t off = 0;
    float* out_adj    = out + off; off += (size_t)N * N;
    float* out_fn     = out + off; off += (size_t)N * Dd;
    float* out_sim    = out + off; off += (size_t)E;
    float* out_pair   = out + off; off += 2 * (size_t)E;
    float* out_before = out + off; off += 1;
    float* out_after  = out + off; off += 1;
    float* out_left   = out + off;

    // Workspace layout
    unsigned* bins       = (unsigned*)d_ws;
    unsigned* offs       = bins + NBINS;
    unsigned* order      = offs + NBINS;
    float*    rowsum     = (float*)(order + E);
    float*    removedsum = rowsum + N;
    unsigned* nnz        = (unsigned*)(removedsum + N);

    const int ZB = (N > NBINS ? N : NBINS);
    zero_ws_kernel<<<(ZB + 255) / 256, 256, 0, stream>>>(bins, offs, removedsum, nnz, N);
    gemm_norm_kernel<<<N / 16, 256, 0, stream>>>(feat, W, b, alpha, out_fn);
    edge_sim_kernel<<<(E * 32 + 255) / 256, 256, 0, stream>>>(pair, out_fn, out_sim, bins, E);
    scan_kernel<<<1, 256, 0, stream>>>(bins);
    scatter_kernel<<<(E + 255) / 256, 256, 0, stream>>>(out_sim, bins, offs, order, E);
    rowsum_kernel<<<N, 256, 0, stream>>>(adj, rowsum, nnz, N);
    removed_kernel<<<(k0 + 255) / 256, 256, 0, stream>>>(order, pair, adj, removedsum, k0, N);
    normalize_kernel<<<16384, 256, 0, stream>>>(adj, rowsum, removedsum, out_adj, N);
    zero_removed_kernel<<<(k0 + 255) / 256, 256, 0, stream>>>(order, pair, out_adj, k0, N);
    finalize_kernel<<<(2 * E + (E - k0) + 1 + 255) / 256, 256, 0, stream>>>(
        pair, order, nnz, out_pair, out_before, out_after, out_left, E, k0, N);
}